// PlainGuidedUpsampler_20607253086407
// MI455X (gfx1250) — compile-verified
//
#include <hip/hip_runtime.h>

typedef __attribute__((ext_vector_type(2))) float v2f;
typedef __attribute__((ext_vector_type(8))) float v8f;

#define WAVES 4          // waves (cells) per block
#define GN    196608     // guidance elements per batch = 3*256*256

// ---------------------------------------------------------------------------
// Kernel 1: deterministic per-batch sum / sum-of-squares of guidance.
// One block per batch (4 blocks), tree reduction in LDS. ws[b]=sum, ws[4+b]=sumsq.
// ---------------------------------------------------------------------------
__global__ __launch_bounds__(256) void jbu_stats(const float* __restrict__ g,
                                                 float* __restrict__ ws) {
  __shared__ float ssum[256];
  __shared__ float ssq[256];
  const int b = blockIdx.x;
  const float* gb = g + (size_t)b * GN;
  float s = 0.f, q = 0.f;
  for (int i = threadIdx.x; i < GN; i += 256) {
    float v = gb[i];
    s += v;
    q += v * v;
  }
  ssum[threadIdx.x] = s;
  ssq[threadIdx.x] = q;
  __syncthreads();
  for (int off = 128; off > 0; off >>= 1) {
    if (threadIdx.x < off) {
      ssum[threadIdx.x] += ssum[threadIdx.x + off];
      ssq[threadIdx.x]  += ssq[threadIdx.x + off];
    }
    __syncthreads();
  }
  if (threadIdx.x == 0) {
    ws[b]     = ssum[0];
    ws[4 + b] = ssq[0];
  }
}

// ---------------------------------------------------------------------------
// Kernel 2: JBU main. One wave32 per 4x4 high-res cell.
// Num[16px,64ch] = W[16,28] x Src[28,64] via v_wmma_f32_16x16x4_f32
// (7 K-chunks x 4 N-tiles, taps 25..27 zero-padded).
// ---------------------------------------------------------------------------
__global__ __launch_bounds__(WAVES * 32) void jbu_main(
    const float* __restrict__ src,   // (4,64,64,64)
    const float* __restrict__ g,     // (4,3,256,256)
    const float* __restrict__ ws,    // [0..3]=sum, [4..7]=sumsq per batch
    float* __restrict__ out) {       // (4,64,256,256)
  __shared__ float4 tapG[WAVES][25];   // low-res guidance (3ch) + spatial dist^2
  __shared__ int    tapOff[WAVES][25]; // source element offset for channel 0
  __shared__ float  denP[WAVES][32];   // per-lane partial denominators

  const int tid  = threadIdx.x;
  const int wave = tid >> 5;
  const int lane = tid & 31;
  const int cell = blockIdx.x * WAVES + wave;   // 0..16383
  const int b  = cell >> 12;
  const int cy = (cell >> 6) & 63;
  const int cx = cell & 63;

  // 1/(2*sigma_range^2): sigma = 0.5*std(ddof=1) -> inv = 2/var
  const float Nf    = (float)GN;
  const float sum   = ws[b];
  const float sumsq = ws[4 + b];
  const float var   = (sumsq - sum * sum / Nf) / (Nf - 1.0f);
  const float inv2sr2 = 2.0f / var;
  const float inv2ss2 = 0.08f;  // 1/(2*2.5^2)

  // Stage the 25 taps' low-res guidance + source offsets into LDS.
  if (lane < 25) {
    const int dy = lane / 5 - 2;
    const int dx = lane % 5 - 2;
    const int yi = min(max(cy + dy, 0), 63);
    const int xi = min(max(cx + dx, 0), 63);
    const int gy = yi * 4 + 2;               // g_low gather: guidance[...,2::4,2::4]
    const int gx = xi * 4 + 2;
    const int gb = (b * 3 * 256 + gy) * 256 + gx;
    float4 t;
    t.x = g[gb];
    t.y = g[gb + 65536];
    t.z = g[gb + 131072];
    t.w = (float)(dy * dy + dx * dx);
    tapG[wave][lane]   = t;
    tapOff[wave][lane] = ((b * 64) * 64 + yi) * 64 + xi;  // + ch*4096 per channel
  }
  __syncthreads();

  // Per-pixel high-res guidance (pixel p = lane&15; both lane halves mirror).
  const int p = lane & 15;
  const int y = cy * 4 + (p >> 2);
  const int x = cx * 4 + (p & 3);
  const int gpix = (b * 3 * 256 + y) * 256 + x;
  const float g0 = g[gpix];
  const float g1 = g[gpix + 65536];
  const float g2 = g[gpix + 131072];

  v8f acc[4] = {};   // 4 N-tiles of 16 channels
  float den = 0.f;
  const int half = (lane >> 4) << 1;  // A/B fragment K split: 0 for lanes<16, 2 else

  #pragma unroll
  for (int kk = 0; kk < 7; ++kk) {
    const int t0 = kk * 4 + half;
    const int t1 = t0 + 1;
    const int tt0 = min(t0, 24);
    const int tt1 = min(t1, 24);
    const float4 T0 = tapG[wave][tt0];
    const float4 T1 = tapG[wave][tt1];
    const float d0 = (g0 - T0.x) * (g0 - T0.x) + (g1 - T0.y) * (g1 - T0.y) +
                     (g2 - T0.z) * (g2 - T0.z);
    const float d1 = (g0 - T1.x) * (g0 - T1.x) + (g1 - T1.y) * (g1 - T1.y) +
                     (g2 - T1.z) * (g2 - T1.z);
    const float w0 = (t0 < 25) ? __expf(-(d0 * inv2sr2 + T0.w * inv2ss2)) : 0.0f;
    const float w1 = (t1 < 25) ? __expf(-(d1 * inv2sr2 + T1.w * inv2ss2)) : 0.0f;
    den += w0 + w1;

    v2f a;
    a.x = w0;   // A: W[p, kbase + (lane<16?0:2)]
    a.y = w1;   // A: W[p, kbase + (lane<16?1:3)]

    const int s0 = tapOff[wave][tt0] + p * 4096;
    const int s1 = tapOff[wave][tt1] + p * 4096;
    #pragma unroll
    for (int nt = 0; nt < 4; ++nt) {
      v2f bb;
      bb.x = src[s0 + nt * 16 * 4096];  // B: Src[kbase+{0|2}, n]
      bb.y = src[s1 + nt * 16 * 4096];  // B: Src[kbase+{1|3}, n]
      acc[nt] = __builtin_amdgcn_wmma_f32_16x16x4_f32(
          false, a, false, bb, (short)0, acc[nt], false, false);
    }
  }

  // Fold denominator across lane halves via LDS (pixel p total = lane p + lane p+16).
  denP[wave][lane] = den;
  __syncthreads();

  const int hi8 = (lane >> 4) << 3;   // D rows: M = r + (lane<16 ? 0 : 8)
  float rden[8];
  #pragma unroll
  for (int r = 0; r < 8; ++r) {
    const int pp = r + hi8;
    rden[r] = 1.0f / (denP[wave][pp] + denP[wave][pp + 16] + 1e-8f);
  }

  #pragma unroll
  for (int nt = 0; nt < 4; ++nt) {
    const int ch = nt * 16 + p;       // D column N = lane&15
    const size_t obase = (size_t)(b * 64 + ch) * 65536;
    #pragma unroll
    for (int r = 0; r < 8; ++r) {
      const int pp = r + hi8;
      const int yy = cy * 4 + (pp >> 2);
      const int xx = cx * 4 + (pp & 3);
      out[obase + yy * 256 + xx] = acc[nt][r] * rden[r];
    }
  }
}

// ---------------------------------------------------------------------------
extern "C" void kernel_launch(void* const* d_in, const int* in_sizes, int n_in,
                              void* d_out, int out_size, void* d_ws, size_t ws_size,
                              hipStream_t stream) {
  const float* src = (const float*)d_in[0];   // source (4,64,64,64)
  const float* g   = (const float*)d_in[1];   // guidance (4,3,256,256)
  float* out = (float*)d_out;                 // (4,64,256,256)
  float* ws  = (float*)d_ws;                  // 8 floats: per-batch sum / sumsq

  jbu_stats<<<4, 256, 0, stream>>>(g, ws);
  // 16384 cells / 4 waves per block
  jbu_main<<<4096, WAVES * 32, 0, stream>>>(src, g, ws, out);
}